// GNBlock_19868518711954
// MI455X (gfx1250) — compile-verified
//
#include <hip/hip_runtime.h>
#include <hip/hip_bf16.h>

#define C_IN  64
#define C_HID 128
#define C_OUT 64
#define BN_EPS 1e-5f

// padded LDS strides (64 banks of 4B; stride % 64 != 0 avoids row-broadcast conflicts)
#define SX_STRIDE 65    // X tile rows (16 x C_IN)
#define SH_STRIDE 129   // H1 tile rows (16 x C_HID)

typedef __attribute__((ext_vector_type(2))) float v2f;
typedef __attribute__((ext_vector_type(8))) float v8f;

// ---------------------------------------------------------------------------
// Kernel 1: per-node fused MLP + root transform using fp32 WMMA.
//   msg   = relu(X@W1 + b1) @ W2 + b2        (written to ws)
//   hbase = X@W_root + b_root                (accumulator base, written to ws)
// One 16-row node tile per block; 8 waves.
// ---------------------------------------------------------------------------
__global__ __launch_bounds__(256) void node_mlp_kernel(
    const float* __restrict__ x,
    const float* __restrict__ W1, const float* __restrict__ b1,
    const float* __restrict__ W2, const float* __restrict__ b2,
    const float* __restrict__ Wr, const float* __restrict__ br,
    float* __restrict__ msg, float* __restrict__ hbase,
    float* __restrict__ colstats, int n_nodes)
{
    __shared__ float sX[16 * SX_STRIDE];   // X tile, 16 x 64 (padded)
    __shared__ float sH1[16 * SH_STRIDE];  // relu(X@W1+b1) tile, 16 x 128 (padded)

    // zero the BN stat accumulators once per launch (ws is poisoned)
    if (blockIdx.x == 0 && threadIdx.x < 2 * C_OUT) colstats[threadIdx.x] = 0.0f;

    const int row0 = blockIdx.x * 16;
    const int tid  = threadIdx.x;
    const int lane = tid & 31;
    const int wave = tid >> 5;
    const int half = lane >> 4;  // 0: lanes 0-15, 1: lanes 16-31
    const int l16  = lane & 15;

    // cooperative load of the 16 x C_IN X tile
    for (int i = tid; i < 16 * C_IN; i += 256) {
        int r = i >> 6, c = i & 63;
        int gr = row0 + r;
        sX[r * SX_STRIDE + c] = (gr < n_nodes) ? x[(size_t)gr * C_IN + c] : 0.0f;
    }
    __syncthreads();

    // ---- Phase 1: H1 = relu(X@W1 + b1); wave w owns columns [16w, 16w+16) ----
    {
        const int col0 = wave * 16;
        v8f c = {};
        #pragma unroll
        for (int k0 = 0; k0 < C_IN; k0 += 4) {
            const int ka = k0 + 2 * half;  // ISA A layout: vgpr0/1 = K 0,1 (lo lanes) / 2,3 (hi lanes)
            v2f a, b;
            a.x = sX[l16 * SX_STRIDE + ka + 0];
            a.y = sX[l16 * SX_STRIDE + ka + 1];
            b.x = W1[(size_t)(ka + 0) * C_HID + col0 + l16];
            b.y = W1[(size_t)(ka + 1) * C_HID + col0 + l16];
            c = __builtin_amdgcn_wmma_f32_16x16x4_f32(false, a, false, b,
                                                      (short)0, c, false, false);
        }
        const float bias = b1[col0 + l16];
        #pragma unroll
        for (int r = 0; r < 8; ++r) {  // C layout: vgpr r -> row r (lo lanes) / r+8 (hi lanes)
            float v = c[r] + bias;
            v = v > 0.0f ? v : 0.0f;   // relu
            sH1[(r + 8 * half) * SH_STRIDE + col0 + l16] = v;
        }
    }
    __syncthreads();

    if (wave < 4) {
        // ---- Phase 2a (waves 0-3): msg = H1@W2 + b2; wave owns cols [16w, 16w+16) ----
        const int col0 = wave * 16;
        v8f c = {};
        #pragma unroll
        for (int k0 = 0; k0 < C_HID; k0 += 4) {
            const int ka = k0 + 2 * half;
            v2f a, b;
            a.x = sH1[l16 * SH_STRIDE + ka + 0];
            a.y = sH1[l16 * SH_STRIDE + ka + 1];
            b.x = W2[(size_t)(ka + 0) * C_OUT + col0 + l16];
            b.y = W2[(size_t)(ka + 1) * C_OUT + col0 + l16];
            c = __builtin_amdgcn_wmma_f32_16x16x4_f32(false, a, false, b,
                                                      (short)0, c, false, false);
        }
        const float bias = b2[col0 + l16];
        #pragma unroll
        for (int r = 0; r < 8; ++r) {
            int gr = row0 + r + 8 * half;
            if (gr < n_nodes) msg[(size_t)gr * C_OUT + col0 + l16] = c[r] + bias;
        }
    } else {
        // ---- Phase 2b (waves 4-7): hbase = X@W_root + b_root ----
        const int col0 = (wave - 4) * 16;
        v8f c = {};
        #pragma unroll
        for (int k0 = 0; k0 < C_IN; k0 += 4) {
            const int ka = k0 + 2 * half;
            v2f a, b;
            a.x = sX[l16 * SX_STRIDE + ka + 0];
            a.y = sX[l16 * SX_STRIDE + ka + 1];
            b.x = Wr[(size_t)(ka + 0) * C_OUT + col0 + l16];
            b.y = Wr[(size_t)(ka + 1) * C_OUT + col0 + l16];
            c = __builtin_amdgcn_wmma_f32_16x16x4_f32(false, a, false, b,
                                                      (short)0, c, false, false);
        }
        const float bias = br[col0 + l16];
        #pragma unroll
        for (int r = 0; r < 8; ++r) {
            int gr = row0 + r + 8 * half;
            if (gr < n_nodes) hbase[(size_t)gr * C_OUT + col0 + l16] = c[r] + bias;
        }
    }
}

// ---------------------------------------------------------------------------
// Kernel 2: edge gather/scatter. One wave per edge; each lane moves a float2
// of the 64-float message with two global f32 atomic adds.
// ---------------------------------------------------------------------------
__global__ __launch_bounds__(256) void edge_scatter_kernel(
    const long long* __restrict__ src, const long long* __restrict__ dst,
    const float* __restrict__ msg, float* __restrict__ h, int n_edges)
{
    const int edge = (int)((blockIdx.x * (unsigned)blockDim.x + threadIdx.x) >> 5);
    const int lane = threadIdx.x & 31;
    if (edge >= n_edges) return;
    const int s = (int)src[edge];
    const int d = (int)dst[edge];
    const float2 m = ((const float2*)(msg + (size_t)s * C_OUT))[lane];
    float* hp = h + (size_t)d * C_OUT + lane * 2;
    atomicAdd(hp + 0, m.x);
    atomicAdd(hp + 1, m.y);
}

// ---------------------------------------------------------------------------
// Kernel 3: PReLU in place + per-column sum / sum-of-squares for BatchNorm.
// Block covers 64 rows; thread (rq, c) strides rows by 4.
// ---------------------------------------------------------------------------
__global__ __launch_bounds__(256) void prelu_stats_kernel(
    float* __restrict__ h, const float* __restrict__ prelu_w,
    float* __restrict__ colstats, int n_nodes)
{
    __shared__ float s_sum[4][C_OUT];
    __shared__ float s_sq[4][C_OUT];
    const float pw = prelu_w[0];
    const int c  = threadIdx.x & 63;
    const int rq = threadIdx.x >> 6;  // 0..3
    const int row0 = blockIdx.x * 64;
    const int rend = min(row0 + 64, n_nodes);
    float sum = 0.0f, sq = 0.0f;
    for (int r = row0 + rq; r < rend; r += 4) {
        float v = h[(size_t)r * C_OUT + c];
        v = v >= 0.0f ? v : pw * v;
        h[(size_t)r * C_OUT + c] = v;
        sum += v;
        sq  += v * v;
    }
    s_sum[rq][c] = sum;
    s_sq[rq][c]  = sq;
    __syncthreads();
    if (rq == 0) {
        sum = s_sum[0][c] + s_sum[1][c] + s_sum[2][c] + s_sum[3][c];
        sq  = s_sq[0][c]  + s_sq[1][c]  + s_sq[2][c]  + s_sq[3][c];
        atomicAdd(&colstats[c], sum);
        atomicAdd(&colstats[C_OUT + c], sq);
    }
}

// ---------------------------------------------------------------------------
// Kernel 4: finalize BatchNorm. Each block rederives scale/shift (64 floats
// of redundant work) and normalizes its slice of h into d_out.
// ---------------------------------------------------------------------------
__global__ __launch_bounds__(256) void bn_kernel(
    const float* __restrict__ h, const float* __restrict__ colstats,
    const float* __restrict__ gamma, const float* __restrict__ beta,
    float* __restrict__ out, int n_nodes)
{
    __shared__ float s_scale[C_OUT];
    __shared__ float s_shift[C_OUT];
    if (threadIdx.x < C_OUT) {
        const int c = threadIdx.x;
        const float invN = 1.0f / (float)n_nodes;
        const float mu   = colstats[c] * invN;
        const float var  = colstats[C_OUT + c] * invN - mu * mu;  // biased var
        const float inv  = rsqrtf(var + BN_EPS);
        const float sc   = inv * gamma[c];
        s_scale[c] = sc;
        s_shift[c] = beta[c] - mu * sc;
    }
    __syncthreads();
    const size_t idx = (size_t)blockIdx.x * blockDim.x + threadIdx.x;
    const size_t total = (size_t)n_nodes * C_OUT;
    if (idx < total) {
        const int c = (int)(idx & 63);
        out[idx] = h[idx] * s_scale[c] + s_shift[c];
    }
}

// ---------------------------------------------------------------------------
extern "C" void kernel_launch(void* const* d_in, const int* in_sizes, int n_in,
                              void* d_out, int out_size, void* d_ws, size_t ws_size,
                              hipStream_t stream) {
    const float*     x     = (const float*)d_in[0];
    const long long* ei    = (const long long*)d_in[1];  // int64 per reference
    const float*     W1    = (const float*)d_in[2];
    const float*     b1    = (const float*)d_in[3];
    const float*     W2    = (const float*)d_in[4];
    const float*     b2    = (const float*)d_in[5];
    const float*     Wr    = (const float*)d_in[6];
    const float*     br    = (const float*)d_in[7];
    const float*     pw    = (const float*)d_in[8];
    const float*     gamma = (const float*)d_in[9];
    const float*     beta  = (const float*)d_in[10];
    float*           out   = (float*)d_out;

    const int n_nodes = in_sizes[0] / C_IN;   // 50000
    const int n_edges = in_sizes[1] / 2;      // 800000

    // workspace layout: [msg N*64][h N*64][colstats 128]
    float* ws       = (float*)d_ws;
    float* msg      = ws;
    float* h        = ws + (size_t)n_nodes * C_OUT;
    float* colstats = ws + (size_t)2 * n_nodes * C_OUT;

    const long long* src = ei;
    const long long* dst = ei + n_edges;

    const int tiles = (n_nodes + 15) / 16;
    node_mlp_kernel<<<tiles, 256, 0, stream>>>(x, W1, b1, W2, b2, Wr, br,
                                               msg, h, colstats, n_nodes);

    const int eblocks = (n_edges + 7) / 8;  // 8 waves (edges) per 256-thread block
    edge_scatter_kernel<<<eblocks, 256, 0, stream>>>(src, dst, msg, h, n_edges);

    const int sblocks = (n_nodes + 63) / 64;
    prelu_stats_kernel<<<sblocks, 256, 0, stream>>>(h, pw, colstats, n_nodes);

    const int nblocks = (int)(((size_t)n_nodes * C_OUT + 255) / 256);
    bn_kernel<<<nblocks, 256, 0, stream>>>(h, colstats, gamma, beta, out, n_nodes);
}